// GATv2ConvModel_29695403884794
// MI455X (gfx1250) — compile-verified
//
#include <hip/hip_runtime.h>
#include <hip/hip_bf16.h>

typedef __attribute__((ext_vector_type(2))) float v2f;
typedef __attribute__((ext_vector_type(8))) float v8f;

#define CDIM 128
#define SLOPE 0.2f
#define EPSV 1e-16f

// ---------- helpers: monotone float<->uint mapping for atomic segment-max ----------
__device__ __forceinline__ unsigned f2mono(float f) {
    unsigned u = __float_as_uint(f);
    return (u & 0x80000000u) ? ~u : (u | 0x80000000u);
}
__device__ __forceinline__ float mono2f(unsigned u) {
    return (u & 0x80000000u) ? __uint_as_float(u & 0x7fffffffu) : __uint_as_float(~u);
}

// ---------------------------------------------------------------------------------
// GEMM: out[N,128] = h[N,128] @ W[128,128] + b   via V_WMMA_F32_16X16X4_F32
// One wave computes a 16x16 tile; 8 waves/block cover all 8 column tiles.
// grid.x = N/16 (N is a multiple of 16: 50000 = 3125*16).
// A frag (16x4 f32): lanes 0-15 hold M=lane, {K0,K1}; lanes 16-31 hold M=lane-16, {K2,K3}.
// B frag (4x16 f32): mirrored — v0/v1 = rows K0/K1 (lanes 0-15), K2/K3 (lanes 16-31).
// C/D (16x16 f32): vgpr r -> row r (lanes 0-15) / row r+8 (lanes 16-31), col = lane&15.
// ---------------------------------------------------------------------------------
__global__ __launch_bounds__(256) void gemm_proj(const float* __restrict__ h,
                                                 const float* __restrict__ W,
                                                 const float* __restrict__ bvec,
                                                 float* __restrict__ out) {
    const int lane  = threadIdx.x & 31;
    const int wave  = threadIdx.x >> 5;      // 0..7 -> column tile
    const int mbase = blockIdx.x << 4;
    const int nbase = wave << 4;
    const int mrow  = mbase + (lane & 15);
    const int ncol  = nbase + (lane & 15);
    const int khalf = (lane >> 4) << 1;      // 0 for lanes 0-15, 2 for lanes 16-31

    v8f c = {0.f, 0.f, 0.f, 0.f, 0.f, 0.f, 0.f, 0.f};

    const float* arow = h + (size_t)mrow * CDIM;
#pragma unroll 4
    for (int k = 0; k < CDIM; k += 4) {
        v2f a = *(const v2f*)(arow + k + khalf);          // 8B-aligned (even index)
        v2f b;
        b.x = W[(size_t)(k + khalf) * CDIM + ncol];
        b.y = W[(size_t)(k + khalf + 1) * CDIM + ncol];
        c = __builtin_amdgcn_wmma_f32_16x16x4_f32(false, a, false, b,
                                                  (short)0, c, false, false);
    }

    const float bb = bvec[ncol];
    const int rbase = mbase + ((lane >> 4) << 3);
#pragma unroll
    for (int r = 0; r < 8; ++r) {
        out[(size_t)(rbase + r) * CDIM + ncol] = c[r] + bb;
    }
}

// ---------------------------------------------------------------------------------
// Per-node init: segment-max (mono-uint encoding, 0 == smallest) and denominator.
// ---------------------------------------------------------------------------------
__global__ void init_nodes(unsigned* __restrict__ mmax, float* __restrict__ den, int n) {
    int i = blockIdx.x * blockDim.x + threadIdx.x;
    if (i < n) { mmax[i] = 0u; den[i] = 0.0f; }
}

// Output init: broadcast bias so the scatter-add accumulates on top of it.
__global__ void init_out_bias(float* __restrict__ out, const float* __restrict__ bias, int n) {
    int i = blockIdx.x * blockDim.x + threadIdx.x;
    if (i < n) out[i] = bias[i & (CDIM - 1)];
}

// ---------------------------------------------------------------------------------
// Edge logits: one wave32 per edge (4 channels/lane), leaky-relu + att dot,
// wave reduction, atomic segment-max per destination node.
// Self-loop edges (e >= Eraw): src=dst=e-Eraw, edge_attr = 1.
// ---------------------------------------------------------------------------------
__global__ __launch_bounds__(256) void edge_logits(const int* __restrict__ ei,
                                                   const float* __restrict__ eattr,
                                                   const float* __restrict__ xl,
                                                   const float* __restrict__ xr,
                                                   const float* __restrict__ We,
                                                   const float* __restrict__ att,
                                                   float* __restrict__ logit,
                                                   unsigned* __restrict__ mmax,
                                                   int Etot, int Eraw) {
    const int e = blockIdx.x * 8 + (threadIdx.x >> 5);
    if (e >= Etot) return;
    const int lane = threadIdx.x & 31;

    int s, d; float ea;
    if (e < Eraw) { s = ei[e]; d = ei[Eraw + e]; ea = eattr[e]; }
    else          { s = d = e - Eraw; ea = 1.0f; }

    const int c0 = lane << 2;
    float4 a  = *(const float4*)(xl + (size_t)s * CDIM + c0);
    float4 b  = *(const float4*)(xr + (size_t)d * CDIM + c0);
    float4 w  = *(const float4*)(We + c0);
    float4 at = *(const float4*)(att + c0);

    float sum = 0.f, v;
    v = a.x + b.x + ea * w.x; v = (v >= 0.f) ? v : SLOPE * v; sum += v * at.x;
    v = a.y + b.y + ea * w.y; v = (v >= 0.f) ? v : SLOPE * v; sum += v * at.y;
    v = a.z + b.z + ea * w.z; v = (v >= 0.f) ? v : SLOPE * v; sum += v * at.z;
    v = a.w + b.w + ea * w.w; v = (v >= 0.f) ? v : SLOPE * v; sum += v * at.w;

#pragma unroll
    for (int off = 16; off > 0; off >>= 1) sum += __shfl_xor(sum, off, 32);

    if (lane == 0) {
        logit[e] = sum;
        atomicMax(mmax + d, f2mono(sum));
    }
}

// ---------------------------------------------------------------------------------
// exp(logit - max[dst]) written in place over logit; atomicAdd into denominator.
// ---------------------------------------------------------------------------------
__global__ void edge_exp(const int* __restrict__ ei,
                         float* __restrict__ logit_ex,
                         const unsigned* __restrict__ mmax,
                         float* __restrict__ den,
                         int Etot, int Eraw) {
    int e = blockIdx.x * blockDim.x + threadIdx.x;
    if (e >= Etot) return;
    int d = (e < Eraw) ? ei[Eraw + e] : (e - Eraw);
    float x = __expf(logit_ex[e] - mono2f(mmax[d]));
    logit_ex[e] = x;
    atomicAdd(den + d, x);
}

// ---------------------------------------------------------------------------------
// Aggregation: out[dst] += alpha * xl[src]; one wave per edge, float atomics.
// ---------------------------------------------------------------------------------
__global__ __launch_bounds__(256) void aggregate(const int* __restrict__ ei,
                                                 const float* __restrict__ xl,
                                                 const float* __restrict__ ex,
                                                 const float* __restrict__ den,
                                                 float* __restrict__ out,
                                                 int Etot, int Eraw) {
    const int e = blockIdx.x * 8 + (threadIdx.x >> 5);
    if (e >= Etot) return;
    const int lane = threadIdx.x & 31;

    int s, d;
    if (e < Eraw) { s = ei[e]; d = ei[Eraw + e]; }
    else          { s = d = e - Eraw; }

    const float alpha = ex[e] / (den[d] + EPSV);
    const int c0 = lane << 2;
    float4 a = *(const float4*)(xl + (size_t)s * CDIM + c0);
    float* o = out + (size_t)d * CDIM + c0;
    atomicAdd(o + 0, a.x * alpha);
    atomicAdd(o + 1, a.y * alpha);
    atomicAdd(o + 2, a.z * alpha);
    atomicAdd(o + 3, a.w * alpha);
}

// ---------------------------------------------------------------------------------
// Host-side orchestration
// ---------------------------------------------------------------------------------
static void run_layer(const float* h, const float* Wl, const float* bl,
                      const float* Wr, const float* br, const float* We,
                      const float* att, const float* bias, float* out,
                      float* xl, float* xr, float* logit_ex,
                      unsigned* mmax, float* den,
                      const int* ei, const float* eattr,
                      int N, int Eraw, int Etot, hipStream_t stream) {
    const int mtiles = N / 16;
    gemm_proj<<<mtiles, 256, 0, stream>>>(h, Wl, bl, xl);
    gemm_proj<<<mtiles, 256, 0, stream>>>(h, Wr, br, xr);
    init_nodes<<<(N + 255) / 256, 256, 0, stream>>>(mmax, den, N);
    init_out_bias<<<((N * CDIM) + 255) / 256, 256, 0, stream>>>(out, bias, N * CDIM);
    edge_logits<<<(Etot + 7) / 8, 256, 0, stream>>>(ei, eattr, xl, xr, We, att,
                                                    logit_ex, mmax, Etot, Eraw);
    edge_exp<<<(Etot + 255) / 256, 256, 0, stream>>>(ei, logit_ex, mmax, den, Etot, Eraw);
    aggregate<<<(Etot + 7) / 8, 256, 0, stream>>>(ei, xl, logit_ex, den, out, Etot, Eraw);
}

extern "C" void kernel_launch(void* const* d_in, const int* in_sizes, int n_in,
                              void* d_out, int out_size, void* d_ws, size_t ws_size,
                              hipStream_t stream) {
    // inputs: x, edge_index(int32, [2,E] flat), edge_attr, then 2 layers x
    //         (Wl, bl, Wr, br, We, att, bias)
    const float* x     = (const float*)d_in[0];
    const int*   ei    = (const int*)d_in[1];
    const float* eattr = (const float*)d_in[2];

    const int N    = in_sizes[0] / CDIM;   // 50000
    const int Eraw = in_sizes[1] / 2;      // 600000
    const int Etot = Eraw + N;             // edges + self loops

    // workspace carve-up (floats)
    float*    xl       = (float*)d_ws;
    float*    xr       = xl + (size_t)N * CDIM;
    float*    hbuf     = xr + (size_t)N * CDIM;
    float*    logit_ex = hbuf + (size_t)N * CDIM;
    unsigned* mmax     = (unsigned*)(logit_ex + Etot);
    float*    den      = (float*)(mmax + N);

    const float* const* P = (const float* const*)(d_in + 3);
    // layer 0: x -> hbuf
    run_layer(x, P[0], P[1], P[2], P[3], P[4], P[5], P[6], hbuf,
              xl, xr, logit_ex, mmax, den, ei, eattr, N, Eraw, Etot, stream);
    // layer 1: hbuf -> d_out
    run_layer(hbuf, P[7], P[8], P[9], P[10], P[11], P[12], P[13], (float*)d_out,
              xl, xr, logit_ex, mmax, den, ei, eattr, N, Eraw, Etot, stream);
}